// StructLoss_7206955122832
// MI455X (gfx1250) — compile-verified
//
#include <hip/hip_runtime.h>
#include <hip/hip_bf16.h>

typedef __attribute__((ext_vector_type(2))) float v2f;
typedef __attribute__((ext_vector_type(8))) float v8f;

#define EPS2 1e-4f
#define INV25 (1.0f / 25.0f)

// ---------------------------------------------------------------------------
// Per-level fused SSIM-structure loss.
// One wave32 block computes a 16x16 tile of output pixels for one batch image.
// Vertical 5-tap box sums are done with V_WMMA_F32_16X16X4_F32 (banded matrix).
// ---------------------------------------------------------------------------
__global__ __launch_bounds__(32)
void ssim_level_kernel(const float* __restrict__ img1,
                       const float* __restrict__ img2,
                       int H,                       // square images, W == H
                       float* __restrict__ partials)
{
    __shared__ float raw1[20 * 20];
    __shared__ float raw2[20 * 20];
    __shared__ float hsum[5][20 * 16];   // 5 quantities, 20 rows x 16 cols

    const int lane = threadIdx.x;        // 0..31 (wave32)
    const int tx = blockIdx.x;
    const int ty = blockIdx.y;
    const int n  = blockIdx.z;           // batch index

    const int r0 = ty * 16;
    const int c0 = tx * 16;
    const int Hp = H - 4;                // output extent (VALID 5x5)

    const float* p1 = img1 + (size_t)n * H * H;
    const float* p2 = img2 + (size_t)n * H * H;

    // ---- Phase A: load 20x20 raw tiles (clamped; out-of-range outputs are
    // masked later so clamping is only to keep addresses legal) -------------
    for (int idx = lane; idx < 400; idx += 32) {
        int r = idx / 20, c = idx % 20;
        int gr = min(r0 + r, H - 1);
        int gc = min(c0 + c, H - 1);
        float a = p1[gr * H + gc];
        float b = p2[gr * H + gc];
        raw1[idx] = a;
        raw2[idx] = b;
    }
    __syncthreads();

    // ---- Phase B: horizontal 5-tap sums of the 5 moment quantities --------
    // 320 entries, exactly 10 per lane -> no divergence.
    for (int idx = lane; idx < 320; idx += 32) {
        int r = idx >> 4, c = idx & 15;
        float s1 = 0.f, s2 = 0.f, s11 = 0.f, s22 = 0.f, s12 = 0.f;
#pragma unroll
        for (int j = 0; j < 5; ++j) {
            float x = raw1[r * 20 + c + j];
            float y = raw2[r * 20 + c + j];
            s1 += x; s2 += y;
            s11 += x * x; s22 += y * y; s12 += x * y;
        }
        hsum[0][idx] = s1;  hsum[1][idx] = s2;
        hsum[2][idx] = s11; hsum[3][idx] = s22;
        hsum[4][idx] = s12;
    }
    __syncthreads();

    // ---- Phase C: vertical 5-tap via banded WMMA --------------------------
    // D(16x16) = Band(16x20) x Hsum(20x16), K split into 5 chunks of 4.
    // A layout (f32 16x4): lane L, vgpr v -> M = L%16, K = v + 2*(L/16).
    // B layout (f32 4x16): lane L, vgpr v -> N = L%16, K = v + 2*(L/16).
    const int half = lane >> 4;
    const int m    = lane & 15;

    v8f cS1 = {}, cS2 = {}, cS11 = {}, cS22 = {}, cS12 = {};

#pragma unroll
    for (int kk = 0; kk < 5; ++kk) {
        v2f a;
        v2f b0, b1, b2, b3, b4;
#pragma unroll
        for (int v = 0; v < 2; ++v) {
            int K = 4 * kk + v + 2 * half;          // 0..19
            a[v] = (K >= m && K <= m + 4) ? 1.0f : 0.0f;
            int i = K * 16 + m;
            b0[v] = hsum[0][i];
            b1[v] = hsum[1][i];
            b2[v] = hsum[2][i];
            b3[v] = hsum[3][i];
            b4[v] = hsum[4][i];
        }
        cS1  = __builtin_amdgcn_wmma_f32_16x16x4_f32(false, a, false, b0, (short)0, cS1,  false, false);
        cS2  = __builtin_amdgcn_wmma_f32_16x16x4_f32(false, a, false, b1, (short)0, cS2,  false, false);
        cS11 = __builtin_amdgcn_wmma_f32_16x16x4_f32(false, a, false, b2, (short)0, cS11, false, false);
        cS22 = __builtin_amdgcn_wmma_f32_16x16x4_f32(false, a, false, b3, (short)0, cS22, false, false);
        cS12 = __builtin_amdgcn_wmma_f32_16x16x4_f32(false, a, false, b4, (short)0, cS12, false, false);
    }

    // ---- Phase D: pointwise loss + edge masking ---------------------------
    // D layout: lane L, vgpr v -> M = v + 8*(L/16), N = L%16.
    float lsum = 0.f;
#pragma unroll
    for (int v = 0; v < 8; ++v) {
        int row = r0 + v + 8 * half;
        int col = c0 + m;
        float mu1  = cS1[v]  * INV25;
        float mu2  = cS2[v]  * INV25;
        float var1 = cS11[v] * INV25 - mu1 * mu1;
        float var2 = cS22[v] * INV25 - mu2 * mu2;
        float cov  = cS12[v] * INV25 - mu1 * mu2;
        float ia = 1.0f / (sqrtf(fmaxf(var1, 0.f) + EPS2) + EPS2);
        float ib = 1.0f / (sqrtf(fmaxf(var2, 0.f) + EPS2) + EPS2);
        float val = var1 * ia * ia + var2 * ib * ib - 2.0f * cov * ia * ib;
        if (row < Hp && col < Hp)
            lsum += val;
    }

    // ---- Phase E: wave32 reduction, one deterministic partial per block ---
    lsum += __shfl_down(lsum, 16);
    lsum += __shfl_down(lsum, 8);
    lsum += __shfl_down(lsum, 4);
    lsum += __shfl_down(lsum, 2);
    lsum += __shfl_down(lsum, 1);
    if (lane == 0) {
        int pidx = ((n * gridDim.y) + ty) * gridDim.x + tx;
        partials[pidx] = lsum;
    }
}

// ---------------------------------------------------------------------------
// Exact bicubic half-downsample (a=-0.75, align_corners=False, edge clamp),
// both pyramid images in one pass.
// ---------------------------------------------------------------------------
__global__ void bicubic_down2_kernel(const float* __restrict__ in1,
                                     const float* __restrict__ in2,
                                     float* __restrict__ out1,
                                     float* __restrict__ out2,
                                     int H)
{
    const float w[4] = { -0.09375f, 0.59375f, 0.59375f, -0.09375f };
    int Ho = H >> 1;
    int tot = 16 * Ho * Ho;
    int t = blockIdx.x * blockDim.x + threadIdx.x;
    if (t >= tot) return;
    int x = t % Ho;
    int y = (t / Ho) % Ho;
    int n = t / (Ho * Ho);
    size_t base = (size_t)n * H * H;
    float acc1 = 0.f, acc2 = 0.f;
#pragma unroll
    for (int j = 0; j < 4; ++j) {
        int ry = min(max(2 * y - 1 + j, 0), H - 1);
        size_t ro = base + (size_t)ry * H;
        float r1 = 0.f, r2 = 0.f;
#pragma unroll
        for (int k = 0; k < 4; ++k) {
            int rx = min(max(2 * x - 1 + k, 0), H - 1);
            r1 += w[k] * in1[ro + rx];
            r2 += w[k] * in2[ro + rx];
        }
        acc1 += w[j] * r1;
        acc2 += w[j] * r2;
    }
    size_t o = (size_t)n * Ho * Ho + (size_t)y * Ho + x;
    out1[o] = acc1;
    out2[o] = acc2;
}

// ---------------------------------------------------------------------------
// Deterministic final reduction: fixed launch config + fixed order per level.
// ---------------------------------------------------------------------------
__global__ __launch_bounds__(256)
void finalize_kernel(const float* __restrict__ partials,
                     const float* __restrict__ weights,
                     float* __restrict__ out)
{
    __shared__ float sdata[256];
    const int counts[4] = { 4096, 1024, 256, 64 };
    const int offs[4]   = { 0, 4096, 5120, 5376 };
    const float denom[4] = { 16.f * 252.f * 252.f,
                             16.f * 124.f * 124.f,
                             16.f *  60.f *  60.f,
                             16.f *  28.f *  28.f };
    float total = 0.f;
    for (int l = 0; l < 4; ++l) {
        float s = 0.f;
        for (int i = threadIdx.x; i < counts[l]; i += 256)
            s += partials[offs[l] + i];
        sdata[threadIdx.x] = s;
        __syncthreads();
        for (int st = 128; st > 0; st >>= 1) {
            if (threadIdx.x < st) sdata[threadIdx.x] += sdata[threadIdx.x + st];
            __syncthreads();
        }
        if (threadIdx.x == 0) total += weights[l] * sdata[0] / denom[l];
        __syncthreads();
    }
    if (threadIdx.x == 0) out[0] = total;
}

// ---------------------------------------------------------------------------
extern "C" void kernel_launch(void* const* d_in, const int* in_sizes, int n_in,
                              void* d_out, int out_size, void* d_ws, size_t ws_size,
                              hipStream_t stream)
{
    const float* fake = (const float*)d_in[0];
    // d_in[1] (hdr_input_original_gray_norm) is unused by the reference.
    const float* hdr  = (const float*)d_in[2];
    const float* wts  = (const float*)d_in[3];
    float* out = (float*)d_out;
    float* ws  = (float*)d_ws;

    // Workspace layout (floats):
    float* partials = ws;                               // 5440
    float* f1 = partials + 5440;                        // 16*128*128
    float* h1 = f1 + 16 * 128 * 128;
    float* f2 = h1 + 16 * 128 * 128;                    // 16*64*64
    float* h2 = f2 + 16 * 64 * 64;
    float* f3 = h2 + 16 * 64 * 64;                      // 16*32*32
    float* h3 = f3 + 16 * 32 * 32;

    // Level 0 (256x256)
    ssim_level_kernel<<<dim3(16, 16, 16), 32, 0, stream>>>(fake, hdr, 256, partials + 0);
    {
        int tot = 16 * 128 * 128;
        bicubic_down2_kernel<<<(tot + 255) / 256, 256, 0, stream>>>(fake, hdr, f1, h1, 256);
    }
    // Level 1 (128x128)
    ssim_level_kernel<<<dim3(8, 8, 16), 32, 0, stream>>>(f1, h1, 128, partials + 4096);
    {
        int tot = 16 * 64 * 64;
        bicubic_down2_kernel<<<(tot + 255) / 256, 256, 0, stream>>>(f1, h1, f2, h2, 128);
    }
    // Level 2 (64x64)
    ssim_level_kernel<<<dim3(4, 4, 16), 32, 0, stream>>>(f2, h2, 64, partials + 5120);
    {
        int tot = 16 * 32 * 32;
        bicubic_down2_kernel<<<(tot + 255) / 256, 256, 0, stream>>>(f2, h2, f3, h3, 64);
    }
    // Level 3 (32x32)
    ssim_level_kernel<<<dim3(2, 2, 16), 32, 0, stream>>>(f3, h3, 32, partials + 5376);

    finalize_kernel<<<1, 256, 0, stream>>>(partials, wts, out);
}